// PatchSupCLRLoss_80324478370482
// MI455X (gfx1250) — compile-verified
//
#include <hip/hip_runtime.h>
#include <hip/hip_bf16.h>

typedef float v2f __attribute__((ext_vector_type(2)));
typedef float v4f __attribute__((ext_vector_type(4)));
typedef float v8f __attribute__((ext_vector_type(8)));
typedef int   v4i __attribute__((ext_vector_type(4)));

#define NN    8192
#define NBLK  7                 // 7 * 16 = 112 >= 100 classes
#define LOG2E 1.4426950408889634f
#define LN2   0.6931471805599453f
#define INV_T (1.0f / 0.07f)

// ---------------------------------------------------------------------------
// Kernel 1: min/max of k  ->  mm[0]=kmax, mm[1]=kmin
// ---------------------------------------------------------------------------
__global__ __launch_bounds__(256) void k_minmax(const float* __restrict__ k,
                                                float* __restrict__ mm) {
    __shared__ float smx[256], smn[256];
    const int tid = threadIdx.x;
    float mx = -__builtin_inff(), mn = __builtin_inff();
    for (int i = tid; i < NN; i += 256) {
        float v = k[i];
        mx = fmaxf(mx, v);
        mn = fminf(mn, v);
    }
    smx[tid] = mx; smn[tid] = mn;
    __syncthreads();
    for (int s = 128; s > 0; s >>= 1) {
        if (tid < s) {
            smx[tid] = fmaxf(smx[tid], smx[tid + s]);
            smn[tid] = fminf(smn[tid], smn[tid + s]);
        }
        __syncthreads();
    }
    if (tid == 0) { mm[0] = smx[0]; mm[1] = smn[0]; }
}

// ---------------------------------------------------------------------------
// Kernel 2: per-row masked softmax-sum via WMMA f32 16x16x4 against one-hot
// labels.  Each wave owns a 16-row tile; 7 accumulators cover 112 class cols.
// row_loss[i] = log( sum_{j: lab_j==lab_i} exp(q_i*k_j/T - m_i) )
// ---------------------------------------------------------------------------
__global__ __launch_bounds__(256) void k_supclr(const float* __restrict__ q,
                                                const float* __restrict__ k,
                                                const int*   __restrict__ labels,
                                                const float* __restrict__ mm,
                                                float* __restrict__ row_loss) {
    __shared__ float lds[8 * NBLK * 256];       // 56 KB: per-wave C spill area

    const int  lane = threadIdx.x & 31;
    const int  wave = __builtin_amdgcn_readfirstlane((int)(threadIdx.x >> 5));
    const int  i0   = (blockIdx.x * 8 + wave) * 16;   // first row of this tile
    const int  m    = lane & 15;                       // row (A) / col (B) id
    const bool hi   = lane >= 16;                      // K=2,3 half of the wave

    const float kmax = mm[0];
    const float kmin = mm[1];

    // Per-lane row constants (duplicated across the two half-waves).
    // exp(q*k/T - m_row) = exp2(qs*k - ms), with the exact row max folded in.
    const float qm = q[i0 + m];
    const float qs = qm * (INV_T * LOG2E);
    const float ms = qs * (qm >= 0.0f ? kmax : kmin);

    v8f C[NBLK];
    const v8f zero = {};
#pragma unroll
    for (int t = 0; t < NBLK; ++t) C[t] = zero;

    for (int j0 = 0; j0 < NN; j0 += 4) {
        // 4 columns per WMMA step; data is wave-uniform.
        const v4f kv = *(const v4f*)(k + j0);
        const v4i lv = *(const v4i*)(labels + j0);
        const int l0 = __builtin_amdgcn_readfirstlane(lv.x);
        const int l1 = __builtin_amdgcn_readfirstlane(lv.y);
        const int l2 = __builtin_amdgcn_readfirstlane(lv.z);
        const int l3 = __builtin_amdgcn_readfirstlane(lv.w);

        // A operand: 16x4 f32, lanes 0-15 hold K=0,1 / lanes 16-31 hold K=2,3
        const float ka = hi ? kv.z : kv.x;
        const float kb = hi ? kv.w : kv.y;
        v2f A;
        A.x = __builtin_amdgcn_exp2f(fmaf(qs, ka, -ms));
        A.y = __builtin_amdgcn_exp2f(fmaf(qs, kb, -ms));

        // B operand: 4x16 one-hot slice of the label matrix (same lane split)
        const int la = hi ? l2 : l0;
        const int lb = hi ? l3 : l1;
        const int na = la - m;                 // == 16*t  <=>  la == 16*t + m
        const int nb = lb - m;

        // Scalar mask of which 16-class blocks the 4 labels touch; skipping
        // is a uniform s_cbranch, so EXEC stays all-ones for WMMA.
        const int blkmask = (1 << (l0 >> 4)) | (1 << (l1 >> 4)) |
                            (1 << (l2 >> 4)) | (1 << (l3 >> 4));
#pragma unroll
        for (int t = 0; t < NBLK; ++t) {
            if (blkmask & (1 << t)) {
                v2f B;
                B.x = (na == t * 16) ? 1.0f : 0.0f;
                B.y = (nb == t * 16) ? 1.0f : 0.0f;
                C[t] = __builtin_amdgcn_wmma_f32_16x16x4_f32(
                    /*neg_a=*/false, A, /*neg_b=*/false, B,
                    /*c_mod=*/(short)0, C[t],
                    /*reuse_a=*/false, /*reuse_b=*/false);
            }
        }
    }

    // Spill C tiles (VGPR v, lanes 0-15 -> M=v / lanes 16-31 -> M=v+8, N=lane&15)
    float* wlds = lds + wave * (NBLK * 256);
    const int mb = hi ? 8 : 0;
#pragma unroll
    for (int t = 0; t < NBLK; ++t) {
#pragma unroll
        for (int v = 0; v < 8; ++v) {
            wlds[t * 256 + (mb + v) * 16 + m] = C[t][v];
        }
    }

    // Cross-lane pick of (row, lab_row) element; wave-private LDS region, so
    // program-order DScnt dependencies are sufficient (no barrier needed).
    if (lane < 16) {
        const int i = i0 + lane;
        const int c = labels[i];
        const float S = wlds[(c >> 4) * 256 + lane * 16 + (c & 15)];
        row_loss[i] = __builtin_amdgcn_logf(S) * LN2;   // natural log
    }
}

// ---------------------------------------------------------------------------
// Kernel 3: deterministic mean of the 8192 row losses -> d_out[0]
// ---------------------------------------------------------------------------
__global__ __launch_bounds__(256) void k_mean(const float* __restrict__ rl,
                                              float* __restrict__ out) {
    __shared__ float s[256];
    const int tid = threadIdx.x;
    float a = 0.0f;
    for (int i = tid; i < NN; i += 256) a += rl[i];
    s[tid] = a;
    __syncthreads();
    for (int r = 128; r > 0; r >>= 1) {
        if (tid < r) s[tid] += s[tid + r];
        __syncthreads();
    }
    if (tid == 0) out[0] = s[0] * (1.0f / (float)NN);
}

// ---------------------------------------------------------------------------
extern "C" void kernel_launch(void* const* d_in, const int* in_sizes, int n_in,
                              void* d_out, int out_size, void* d_ws, size_t ws_size,
                              hipStream_t stream) {
    const float* q      = (const float*)d_in[0];   // feat_q [8192]
    const float* k      = (const float*)d_in[1];   // feat_k [8192]
    const int*   labels = (const int*)d_in[2];     // labels [8192]

    float* ws       = (float*)d_ws;
    float* row_loss = ws;          // [0 .. 8191]
    float* mm       = ws + NN;     // [8192]=kmax, [8193]=kmin

    k_minmax<<<1, 256, 0, stream>>>(k, mm);
    k_supclr<<<NN / 16 / 8, 256, 0, stream>>>(q, k, labels, mm, row_loss);
    k_mean<<<1, 256, 0, stream>>>(row_loss, (float*)d_out);
}